// Message_44513041056428
// MI455X (gfx1250) — compile-verified
//
#include <hip/hip_runtime.h>
#include <hip/hip_bf16.h>
#include <math.h>

typedef float v2f __attribute__((ext_vector_type(2)));
typedef float v8f __attribute__((ext_vector_type(8)));
typedef int   v4i __attribute__((ext_vector_type(4)));

#define FDIM 128
#define F3   384
#define NRBF 20
#define RC   5.0f
#define ROWS 48           // rows per block = 3 M-tiles of 16

#if __has_builtin(__builtin_amdgcn_global_load_async_to_lds_b128) && \
    __has_builtin(__builtin_amdgcn_s_wait_asynccnt)
#define HAS_ASYNC_LDS 1
#else
#define HAS_ASYNC_LDS 0
#endif

// Copy 16 bytes global -> LDS. Async direct-to-LDS path on gfx1250 when available.
__device__ __forceinline__ void copy_f4_to_lds(const float* gsrc, float* ldst)
{
#if HAS_ASYNC_LDS
    __builtin_amdgcn_global_load_async_to_lds_b128(
        (__attribute__((address_space(1))) v4i*)gsrc,
        (__attribute__((address_space(3))) v4i*)ldst, 0, 0);
#else
    *(float4*)ldst = *(const float4*)gsrc;
#endif
}

__device__ __forceinline__ void lds_copy_fence()
{
#if HAS_ASYNC_LDS
    __builtin_amdgcn_s_wait_asynccnt(0);
#endif
    __syncthreads();
}

__device__ __forceinline__ float fast_silu(float x)
{
    // x * sigmoid(x) with v_rcp_f32 instead of IEEE division
    return x * __builtin_amdgcn_rcpf(1.0f + __expf(-x));
}

// ---------------------------------------------------------------------------
// Node MLP:  S[A x 384] = silu(X[A x 128] @ W1[128x128] + b1) @ W2[128x384] + b2
// One block per 48 rows, 8 waves. Each wave holds 3 accumulators (3 M-tiles)
// so every B-fragment (weight) load feeds 3 V_WMMA_F32_16X16X4_F32 ops.
// ---------------------------------------------------------------------------
__global__ __launch_bounds__(256) void node_mlp_kernel(
    const float* __restrict__ X,
    const float* __restrict__ W1, const float* __restrict__ b1,
    const float* __restrict__ W2, const float* __restrict__ b2,
    float* __restrict__ S, int A)
{
    __shared__ float xt[ROWS][132];   // stride 132 -> conflict-free (bank = 4*row+k)
    __shared__ float ht[ROWS][132];

    const int tid  = threadIdx.x;
    const int wave = tid >> 5;
    const int lane = tid & 31;
    const int m0   = blockIdx.x * ROWS;

    // stage X tile (48 x 128) in LDS, 16B chunks, async direct-to-LDS
    for (int i = tid; i < ROWS * (FDIM / 4); i += 256) {
        int r = i >> 5;                 // row 0..47
        int c = (i & 31) << 2;          // col 0,4,...,124
        int row = m0 + r; if (row >= A) row = A - 1;
        copy_f4_to_lds(X + (size_t)row * FDIM + c, &xt[r][c]);
    }
    lds_copy_fence();

    const int half = lane >> 4;   // 0: K={0,1}, 1: K={2,3} within each x4 step
    const int l16  = lane & 15;

    // ---- phase 1: h = silu(X @ W1 + b1); wave owns columns [wave*16, +16)
    {
        const int n = wave * 16 + l16;
        const float bv = b1[n];
        v8f acc0 = {bv, bv, bv, bv, bv, bv, bv, bv};
        v8f acc1 = acc0, acc2 = acc0;
        for (int k = 0; k < FDIM; k += 4) {
            const int kk = k + (half << 1);
            v2f b; b.x = W1[kk * FDIM + n]; b.y = W1[(kk + 1) * FDIM + n];
            v2f a0, a1, a2;
            a0.x = xt[l16][kk];      a0.y = xt[l16][kk + 1];
            a1.x = xt[16 + l16][kk]; a1.y = xt[16 + l16][kk + 1];
            a2.x = xt[32 + l16][kk]; a2.y = xt[32 + l16][kk + 1];
            acc0 = __builtin_amdgcn_wmma_f32_16x16x4_f32(false, a0, false, b, (short)0, acc0, false, false);
            acc1 = __builtin_amdgcn_wmma_f32_16x16x4_f32(false, a1, false, b, (short)0, acc1, false, false);
            acc2 = __builtin_amdgcn_wmma_f32_16x16x4_f32(false, a2, false, b, (short)0, acc2, false, false);
        }
#pragma unroll
        for (int v = 0; v < 8; ++v) {
            const int M = v + (half << 3);
            ht[M][n]      = fast_silu(acc0[v]);
            ht[16 + M][n] = fast_silu(acc1[v]);
            ht[32 + M][n] = fast_silu(acc2[v]);
        }
    }
    __syncthreads();

    // ---- phase 2: S = h @ W2 + b2; wave owns 3 column tiles of the 384 outputs
    for (int j = 0; j < 3; ++j) {
        const int n = (wave + 8 * j) * 16 + l16;
        const float bv = b2[n];
        v8f acc0 = {bv, bv, bv, bv, bv, bv, bv, bv};
        v8f acc1 = acc0, acc2 = acc0;
        for (int k = 0; k < FDIM; k += 4) {
            const int kk = k + (half << 1);
            v2f b; b.x = W2[kk * F3 + n]; b.y = W2[(kk + 1) * F3 + n];
            v2f a0, a1, a2;
            a0.x = ht[l16][kk];      a0.y = ht[l16][kk + 1];
            a1.x = ht[16 + l16][kk]; a1.y = ht[16 + l16][kk + 1];
            a2.x = ht[32 + l16][kk]; a2.y = ht[32 + l16][kk + 1];
            acc0 = __builtin_amdgcn_wmma_f32_16x16x4_f32(false, a0, false, b, (short)0, acc0, false, false);
            acc1 = __builtin_amdgcn_wmma_f32_16x16x4_f32(false, a1, false, b, (short)0, acc1, false, false);
            acc2 = __builtin_amdgcn_wmma_f32_16x16x4_f32(false, a2, false, b, (short)0, acc2, false, false);
        }
#pragma unroll
        for (int v = 0; v < 8; ++v) {
            const int M = v + (half << 3);
            const int r0 = m0 + M, r1 = m0 + 16 + M, r2 = m0 + 32 + M;
            if (r0 < A) S[(size_t)r0 * F3 + n] = acc0[v];
            if (r1 < A) S[(size_t)r1 * F3 + n] = acc1[v];
            if (r2 < A) S[(size_t)r2 * F3 + n] = acc2[v];
        }
    }
}

// ---------------------------------------------------------------------------
// Edge kernel: rbf -> r = rbf@Wr+br, cutoff, combine with gathered s[idx_i],
// form delta_s / delta_v, scatter-add into outputs with f32 global atomics.
// One wave per edge, 128 edges per block (Wr staged once per block in LDS).
// ---------------------------------------------------------------------------
__global__ __launch_bounds__(256) void edge_kernel(
    const float* __restrict__ dirs,
    const int* __restrict__ idx_i, const int* __restrict__ idx_j,
    const float* __restrict__ Wr, const float* __restrict__ br,
    const float* __restrict__ S, const float* __restrict__ V,
    float* __restrict__ out_dv, float* __restrict__ out_ds, int E)
{
    __shared__ float wr_s[NRBF * F3];
    __shared__ float br_s[F3];

    const int tid = threadIdx.x;
    for (int i = tid; i < (NRBF * F3) / 4; i += 256)
        copy_f4_to_lds(Wr + 4 * i, &wr_s[4 * i]);
    if (tid < F3 / 4)
        copy_f4_to_lds(br + 4 * tid, &br_s[4 * tid]);
    lds_copy_fence();

    const int wave = tid >> 5;
    const int lane = tid & 31;
    const float K = 3.14159265358979323846f / RC;

    const int ebase = blockIdx.x * 128;
    for (int it = 0; it < 16; ++it) {
        const int e = ebase + it * 8 + wave;
        if (e >= E) break;

        const float dx = dirs[(size_t)e * 3 + 0];
        const float dy = dirs[(size_t)e * 3 + 1];
        const float dz = dirs[(size_t)e * 3 + 2];
        const float n2   = dx * dx + dy * dy + dz * dz;
        const float inv  = __builtin_amdgcn_rsqf(n2);   // 1/|r|
        const float norm = n2 * inv;                    // |r|

        // lane k < 20 computes rbf_k = sin((k+1)*pi/rc * |r|) / |r|
        float sv = 0.0f;
        if (lane < NRBF) sv = __sinf((float)(lane + 1) * K * norm) * inv;

        // cosine cutoff with hard mask (min(delta,0)/delta semantics)
        const float mask = (norm < RC) ? 1.0f : 0.0f;
        const float fc = 0.5f * (__cosf(K * norm) + 1.0f) * mask;

        const int i = idx_i[e];
        const int j = idx_j[e];

        // r[n] = (rbf @ Wr + br)[n]; lane owns n = lane + 32*t, t=0..11
        float r[12];
#pragma unroll
        for (int t = 0; t < 12; ++t) r[t] = br_s[lane + 32 * t];
        for (int k = 0; k < NRBF; ++k) {
            const float w = __shfl(sv, k, 32);
#pragma unroll
            for (int t = 0; t < 12; ++t)
                r[t] += w * wr_s[k * F3 + lane + 32 * t];
        }

        // combined = s[i] * (fc * r)
        const float* si = S + (size_t)i * F3;
#pragma unroll
        for (int t = 0; t < 12; ++t)
            r[t] = fc * r[t] * si[lane + 32 * t];

        // delta_s scatter (first 128 of combined)
#pragma unroll
        for (int q = 0; q < 4; ++q)
            atomicAdd(&out_ds[(size_t)j * FDIM + lane + 32 * q], r[q]);

        // delta_v = vectors[i]*dv1 + dv2 * unit_dir ; scatter
        const float u[3] = {dx * inv, dy * inv, dz * inv};
        const float* vi = V + (size_t)i * 3 * FDIM;
#pragma unroll
        for (int d = 0; d < 3; ++d) {
#pragma unroll
            for (int q = 0; q < 4; ++q) {
                const int f = lane + 32 * q;
                const float dv = vi[d * FDIM + f] * r[4 + q] + r[8 + q] * u[d];
                atomicAdd(&out_dv[((size_t)j * 3 + d) * FDIM + f], dv);
            }
        }
    }
}

extern "C" void kernel_launch(void* const* d_in, const int* in_sizes, int n_in,
                              void* d_out, int out_size, void* d_ws, size_t ws_size,
                              hipStream_t stream)
{
    const float* vectors = (const float*)d_in[0];
    const float* scalars = (const float*)d_in[1];
    const float* dirs    = (const float*)d_in[2];
    const int*   idx_i   = (const int*)d_in[3];
    const int*   idx_j   = (const int*)d_in[4];
    const float* W1      = (const float*)d_in[5];
    const float* b1      = (const float*)d_in[6];
    const float* W2      = (const float*)d_in[7];
    const float* b2      = (const float*)d_in[8];
    const float* Wr      = (const float*)d_in[9];
    const float* br      = (const float*)d_in[10];

    const int A = in_sizes[1] / FDIM;   // scalars: A x 1 x F
    const int E = in_sizes[3];          // idx_i

    float* S      = (float*)d_ws;                       // A x 384 intermediate
    float* out_dv = (float*)d_out;                      // A x 3 x 128 (first)
    float* out_ds = out_dv + (size_t)A * 3 * FDIM;      // A x 1 x 128 (second)

    // atomics accumulate -> zero outputs every call (capture-safe)
    (void)hipMemsetAsync(d_out, 0, (size_t)out_size * sizeof(float), stream);

    const int tiles = (A + ROWS - 1) / ROWS;
    node_mlp_kernel<<<tiles, 256, 0, stream>>>(scalars, W1, b1, W2, b2, S, A);

    const int eblocks = (E + 127) / 128;
    edge_kernel<<<eblocks, 256, 0, stream>>>(dirs, idx_i, idx_j, Wr, br,
                                             S, vectors, out_dv, out_ds, E);
}